// VMDModule_51926154609386
// MI455X (gfx1250) — compile-verified
//
#include <hip/hip_runtime.h>

// ---------------- problem constants (from reference) ----------------
#define B_DIM   16
#define C_DIM   4
#define K_MODES 4
#define T_LEN   8192
#define T2      16384
#define HALFN   8192
#define NSIG    (B_DIM * C_DIM)     // 64 independent signals
#define ALPHA_C 2000.0f
#define N_ITERS 20
#define U_ELEMS (B_DIM * K_MODES * T_LEN * C_DIM)  // 2097152 floats

typedef float v2f __attribute__((ext_vector_type(2)));
typedef float v8f __attribute__((ext_vector_type(8)));
typedef unsigned int u32x4 __attribute__((ext_vector_type(4)));
typedef int i32x4 __attribute__((ext_vector_type(4)));
typedef int i32x8 __attribute__((ext_vector_type(8)));

#if defined(__gfx1250__) && __has_builtin(__builtin_amdgcn_wmma_f32_16x16x4_f32)
#define USE_WMMA 1
#else
#define USE_WMMA 0
#endif

#if defined(__gfx1250__) && __has_builtin(__builtin_amdgcn_tensor_load_to_lds) && \
    __has_builtin(__builtin_amdgcn_s_wait_tensorcnt)
#define USE_TDM 1
#else
#define USE_TDM 0
#endif

struct cpx { float x, y; };
__device__ __forceinline__ cpx cadd(cpx a, cpx b){ return {a.x + b.x, a.y + b.y}; }
__device__ __forceinline__ cpx csub(cpx a, cpx b){ return {a.x - b.x, a.y - b.y}; }
__device__ __forceinline__ cpx cmul(cpx a, cpx b){ return {a.x*b.x - a.y*b.y, a.x*b.y + a.y*b.x}; }

__device__ __forceinline__ int bitrev10(int v) { return (int)(__brev((unsigned)v) >> 22); }

// --------------------------------------------------------------------
// 16384-point complex FFT, four-step: 16384 = 16 (WMMA DFT) x 1024 (DIT).
// Single in-place LDS buffer (128 KB). Caller must pre-load with the input
// permutation buf[n1*1024 + bitrev10(n2)] = x[n1 + 16*n2].
// sign = -1 forward, +1 inverse (unscaled). Result: buf[k] = F[k], natural order.
// Requires blockDim.x == 1024 (32 wave32 waves, EXEC all-ones for WMMA).
// Forced inline so InferAddressSpaces keeps buf in addrspace(3) (direct ds ops).
// --------------------------------------------------------------------
__device__ __forceinline__ void fft16384(cpx* buf, int tid, float sign)
{
    // ---- 16 column FFTs of length 1024: in-place radix-2 DIT, 10 stages
    #pragma unroll 1
    for (int s = 0; s < 10; ++s) {
        const int half = 1 << s;
        const float w0 = sign * 6.2831853071795864f / (float)(half << 1);
        #pragma unroll
        for (int it = 0; it < 8; ++it) {
            int idx  = tid + (it << 10);     // 0..8191 : 16 cols * 512 butterflies
            int col  = idx >> 9;
            int u    = idx & 511;
            int j    = u & (half - 1);       // position inside group
            int base = (u >> s) << (s + 1);  // group base (stride len)
            int i0   = (col << 10) + base + j;
            int i1   = i0 + half;
            float sn, cs;
            __sincosf(w0 * (float)j, &sn, &cs);
            cpx a = buf[i0];
            cpx b = cmul(buf[i1], {cs, sn});
            buf[i0] = cadd(a, b);
            buf[i1] = csub(a, b);
        }
        __syncthreads();
    }
    // buf[n1*1024 + k2] = G[n1][k2]
    // ---- four-step twiddle: H[n1][k2] = G[n1][k2] * exp(sign*2*pi*i*n1*k2/16384)
    const float wN = sign * 6.2831853071795864f / 16384.0f;
    #pragma unroll
    for (int it = 0; it < 16; ++it) {
        int o  = tid + (it << 10);
        int n1 = o >> 10;
        int k2 = o & 1023;
        float sn, cs;
        __sincosf(wN * (float)(n1 * k2), &sn, &cs);
        buf[o] = cmul(buf[o], {cs, sn});
    }
    __syncthreads();
    // ---- final DFT-16 over n1, batched 16 k2-columns per tile:
    //      F[k2 + 1024*k1] = sum_n1 exp(sign*2*pi*i*k1*n1/16) * H[n1][k2]
    // Complex 16x16x16 matmul per tile -> 16 x V_WMMA_F32_16X16X4_F32.
    // In-place safe: tile reads {n1*1024+k2} fully before writing {k1*1024+k2}
    // (same address set); tiles are disjoint across waves.
    const int lane = tid & 31;
    const int wid  = tid >> 5;
    const int hi   = lane >> 4;    // lane group (0: lanes 0-15, 1: lanes 16-31)
    const int M    = lane & 15;    // A row (=k1) / B,C column index per ISA layout
    #pragma unroll 1
    for (int b2 = 0; b2 < 2; ++b2) {           // 64 tiles / 32 waves = 2 per wave
        const int k20 = ((wid << 1) + b2) << 4;
#if USE_WMMA
        v8f yr = {}; v8f yi = {};
        v2f arc[4], aic[4], anc[4], brc[4], bic[4];
        #pragma unroll
        for (int c = 0; c < 4; ++c) {          // gather all reads first (in-place)
            #pragma unroll
            for (int v = 0; v < 2; ++v) {
                // A layout (16x4 f32): element (M, K=v+2*hi) ; chunk offset 4*c
                int n1 = 4 * c + v + 2 * hi;
                float th = 0.39269908169872f * (float)((M * n1) & 15); // 2*pi/16
                float sn, cs;
                __sincosf(sign * th, &sn, &cs);
                arc[c][v] = cs; aic[c][v] = sn; anc[c][v] = -sn;
                // B layout (4x16 f32): row K=v+2*hi of chunk, column M
                cpx h = buf[(n1 << 10) + k20 + M];
                brc[c][v] = h.x; bic[c][v] = h.y;
            }
        }
        #pragma unroll
        for (int c = 0; c < 4; ++c) {
            // Yr += Ar*Br - Ai*Bi ; Yi += Ar*Bi + Ai*Br   (sign folded into ai/an)
            yr = __builtin_amdgcn_wmma_f32_16x16x4_f32(false, arc[c], false, brc[c], (short)0, yr, false, false);
            yr = __builtin_amdgcn_wmma_f32_16x16x4_f32(false, anc[c], false, bic[c], (short)0, yr, false, false);
            yi = __builtin_amdgcn_wmma_f32_16x16x4_f32(false, arc[c], false, bic[c], (short)0, yi, false, false);
            yi = __builtin_amdgcn_wmma_f32_16x16x4_f32(false, aic[c], false, brc[c], (short)0, yi, false, false);
        }
        #pragma unroll
        for (int r = 0; r < 8; ++r) {          // C/D layout: row = r + 8*hi, col = M
            int k1 = r + 8 * hi;
            buf[k20 + M + (k1 << 10)] = {yr[r], yi[r]};
        }
#else
        // scalar fallback with identical output mapping
        float yr[8], yi[8];
        cpx hv[16];
        #pragma unroll
        for (int n1 = 0; n1 < 16; ++n1) hv[n1] = buf[(n1 << 10) + k20 + M];
        #pragma unroll
        for (int r = 0; r < 8; ++r) { yr[r] = 0.f; yi[r] = 0.f; }
        for (int n1 = 0; n1 < 16; ++n1) {
            #pragma unroll
            for (int r = 0; r < 8; ++r) {
                int k1 = r + 8 * hi;
                float th = 0.39269908169872f * (float)((k1 * n1) & 15);
                float sn, cs;
                __sincosf(sign * th, &sn, &cs);
                yr[r] += cs * hv[n1].x - sn * hv[n1].y;
                yi[r] += cs * hv[n1].y + sn * hv[n1].x;
            }
        }
        #pragma unroll
        for (int r = 0; r < 8; ++r) {
            int k1 = r + 8 * hi;
            buf[k20 + M + (k1 << 10)] = {yr[r], yi[r]};
        }
#endif
    }
    __syncthreads();
}

// ------------------- kernel 1: mirror-extend + forward FFT -------------------
// One workgroup per (b,c). The input column x[b, :, c] (1x8192 tile, 16 B row
// stride) is staged into LDS by the Tensor Data Mover, then mirror-extended
// with the DIT bit-reversal folded into the LDS->LDS permutation.
__global__ __launch_bounds__(1024) void vmd_fwd_fft(const float* __restrict__ x,
                                                    cpx* __restrict__ Fp)
{
    __shared__ cpx   buf[T2];                // 128 KB static LDS
    __shared__ float stage[T_LEN];           // 32 KB TDM landing zone
    const int sig = blockIdx.x;              // b*4 + c
    const int b = sig >> 2, c = sig & 3;
    const int tid = threadIdx.x;
    const float* xs = x + (size_t)b * (T_LEN * C_DIM) + c;   // x is (B, T, C)

#if USE_TDM
    if ((tid >> 5) == 0) {                   // wave 0 issues the DMA (EXEC ignored)
        unsigned long long ga = (unsigned long long)(uintptr_t)xs;
        unsigned int lds_off  = (unsigned int)(uintptr_t)(void*)stage; // addr[31:0] = LDS byte offset
        // D# group 0: count=1 | lds_addr | global_addr | type=2 ("image")
        u32x4 g0 = { 1u,
                     lds_off,
                     (unsigned int)(ga & 0xFFFFFFFFu),
                     (unsigned int)((ga >> 32) & 0x01FFFFFFu) | (2u << 30) };
        // D# group 1: data_size=4B(code 2); tensor_dim0=4, tensor_dim1=8192;
        //             tile_dim0=1, tile_dim1=8192; tensor_dim0_stride=4 elems
        i32x8 g1 = { (int)(2u << 16),        // workgroup_mask=0, data_size=2
                     (int)(4u << 16),        // tensor_dim0[15:0] @ bits 63:48
                     (int)(8192u << 16),     // tensor_dim1[15:0] @ bits 95:80
                     (int)(1u << 16),        // tile_dim0=1       @ bits 127:112
                     (int)(8192u),           // tile_dim1=8192    @ bits 143:128
                     (int)(4u),              // tensor_dim0_stride low32
                     0, 0 };
        i32x4 g2 = { 0, 0, 0, 0 };           // 2D tensor: groups 2/3 unused
        i32x4 g3 = { 0, 0, 0, 0 };
        i32x8 g4 = { 0, 0, 0, 0, 0, 0, 0, 0 };
        __builtin_amdgcn_tensor_load_to_lds(g0, g1, g2, g3, g4, 0);
        __builtin_amdgcn_s_wait_tensorcnt(0);
    }
    __syncthreads();                         // publish TDM-written LDS to all waves
#endif

    #pragma unroll
    for (int it = 0; it < 16; ++it) {
        int i = tid + (it << 10);
        int ts;
        if      (i < 4096)  ts = 4095 - i;       // x[:4096] reversed
        else if (i < 12288) ts = i - 4096;       // x
        else                ts = 20479 - i;      // x[4096:] reversed
#if USE_TDM
        float v = stage[ts];
#else
        float v = xs[ts * 4];
#endif
        // decimated layout + DIT bit-reversal folded into the load permutation
        buf[((i & 15) << 10) + bitrev10(i >> 4)] = {v, 0.0f};
    }
    __syncthreads();
    fft16384(buf, tid, -1.0f);
    cpx* out = Fp + (size_t)sig * HALFN;
    #pragma unroll
    for (int it = 0; it < 8; ++it) {
        int j = tid + (it << 10);
        out[j] = buf[j];                  // fftshift: positive half == F[0:8192]
    }
}

// ------------------- kernel 2: 20 ADMM iterations (positive half) ------------
// One workgroup per (b,c); u_hat lives entirely in registers (4 modes x 8 bins).
__global__ __launch_bounds__(1024) void vmd_iter(const cpx* __restrict__ Fp_all,
                                                 cpx* __restrict__ U,
                                                 float* __restrict__ Omega)
{
    __shared__ float redA[32];
    __shared__ float redB[32];
    const int sig = blockIdx.x;
    const int tid = threadIdx.x;
    const cpx* Fp = Fp_all + (size_t)sig * HALFN;

    cpx f[8], u[K_MODES][8], S[8];
    float fr[8], om[K_MODES];
    #pragma unroll
    for (int r = 0; r < 8; ++r) {
        int j = tid + (r << 10);
        f[r]  = Fp[j];
        fr[r] = (float)j * (1.0f / 16384.0f);   // freqs on positive half
        S[r]  = {0.f, 0.f};
        #pragma unroll
        for (int k = 0; k < K_MODES; ++k) u[k][r] = {0.f, 0.f};
    }
    #pragma unroll
    for (int k = 0; k < K_MODES; ++k) om[k] = 0.125f * (float)k;  // 0.5/K * k

    #pragma unroll 1
    for (int itn = 0; itn < N_ITERS; ++itn) {
        #pragma unroll 1
        for (int k = 0; k < K_MODES; ++k) {
            float pa = 0.f, pb = 0.f;
            const float wk = om[k];
            #pragma unroll
            for (int r = 0; r < 8; ++r) {
                S[r] = csub(S[r], u[k][r]);
                float df  = fr[r] - wk;
                float den = 1.0f + ALPHA_C * df * df;
                cpx nu = { (f[r].x - S[r].x) / den, (f[r].y - S[r].y) / den };
                S[r] = cadd(S[r], nu);
                u[k][r] = nu;
                float p = nu.x * nu.x + nu.y * nu.y;
                pa = fmaf(fr[r], p, pa);
                pb += p;
            }
            // block reduction: wave32 shuffle then LDS combine
            #pragma unroll
            for (int off = 16; off > 0; off >>= 1) {
                pa += __shfl_down(pa, off);
                pb += __shfl_down(pb, off);
            }
            if ((tid & 31) == 0) { redA[tid >> 5] = pa; redB[tid >> 5] = pb; }
            __syncthreads();
            float sa = 0.f, sb = 0.f;
            #pragma unroll
            for (int w = 0; w < 32; ++w) { sa += redA[w]; sb += redB[w]; }
            __syncthreads();
            om[k] = sa / sb;
        }
    }
    // store u_hat positive halves + omega
    #pragma unroll
    for (int k = 0; k < K_MODES; ++k) {
        cpx* uo = U + ((size_t)sig * K_MODES + k) * HALFN;
        #pragma unroll
        for (int r = 0; r < 8; ++r) uo[tid + (r << 10)] = u[k][r];
    }
    if (tid == 0) {
        #pragma unroll
        for (int k = 0; k < K_MODES; ++k) Omega[sig * K_MODES + k] = om[k];
    }
}

// ------------------- kernel 3: Hermitian reconstruct + inverse FFT -----------
// One workgroup per (b,c,k). Output u transposed to (B, K, T, C).
__global__ __launch_bounds__(1024) void vmd_inv_fft(const cpx* __restrict__ U,
                                                    float* __restrict__ out)
{
    __shared__ cpx buf[T2];                  // 128 KB static LDS
    const int blk = blockIdx.x;              // (b*4 + c)*4 + k
    const int k   = blk & 3;
    const int sig = blk >> 2;
    const int b   = sig >> 2, c = sig & 3;
    const int tid = threadIdx.x;
    const cpx* pos = U + (size_t)blk * HALFN;

    // Spectrum after ifftshift of the reference's `full`:
    // G[0]=conj(pos[0]); G[n]=pos[n] n=1..8191; G[8192]=conj(pos[8191]);
    // G[m]=conj(pos[16384-m]) m=8193..16383.
    #pragma unroll
    for (int it = 0; it < 16; ++it) {
        int m = tid + (it << 10);
        cpx g;
        if      (m == 0)    { cpx p0 = pos[0];         g = {p0.x, -p0.y}; }
        else if (m < 8192)  {                          g = pos[m];        }
        else if (m == 8192) { cpx pe = pos[8191];      g = {pe.x, -pe.y}; }
        else                { cpx pp = pos[16384 - m]; g = {pp.x, -pp.y}; }
        buf[((m & 15) << 10) + bitrev10(m >> 4)] = g;
    }
    __syncthreads();
    fft16384(buf, tid, +1.0f);
    const float scl = 1.0f / 16384.0f;               // ifft normalization
    float* ob = out + (size_t)b * (K_MODES * T_LEN * C_DIM)
                    + (size_t)k * (T_LEN * C_DIM) + c;
    #pragma unroll
    for (int it = 0; it < 8; ++it) {
        int t = tid + (it << 10);
        ob[(size_t)t * C_DIM] = buf[4096 + t].x * scl;  // real part, middle half
    }
}

// ------------------- kernel 4: omega mean over channels ----------------------
__global__ void vmd_omega_mean(const float* __restrict__ Om, float* __restrict__ ow)
{
    int i = threadIdx.x;
    if (i < B_DIM * K_MODES) {
        int b = i >> 2, k = i & 3;
        float s = 0.f;
        #pragma unroll
        for (int c = 0; c < C_DIM; ++c) s += Om[(b * C_DIM + c) * K_MODES + k];
        ow[i] = 0.25f * s;
    }
}

// -----------------------------------------------------------------------------
extern "C" void kernel_launch(void* const* d_in, const int* in_sizes, int n_in,
                              void* d_out, int out_size, void* d_ws, size_t ws_size,
                              hipStream_t stream)
{
    (void)in_sizes; (void)n_in; (void)out_size; (void)ws_size;
    const float* x = (const float*)d_in[0];
    float* out = (float*)d_out;

    char* ws = (char*)d_ws;
    cpx*  Fp = (cpx*)ws;                                         // 64*8192 cpx (4 MB)
    cpx*  U  = (cpx*)(ws + (size_t)NSIG * HALFN * sizeof(cpx));  // 64*4*8192 cpx (16 MB)
    float* Om = (float*)(ws + (size_t)NSIG * HALFN * sizeof(cpx)
                            + (size_t)NSIG * K_MODES * HALFN * sizeof(cpx));

    vmd_fwd_fft <<<NSIG,           1024, 0, stream>>>(x, Fp);
    vmd_iter    <<<NSIG,           1024, 0, stream>>>(Fp, U, Om);
    vmd_inv_fft <<<NSIG * K_MODES, 1024, 0, stream>>>(U, out);
    vmd_omega_mean<<<1, 64, 0, stream>>>(Om, out + U_ELEMS);
}